// TransformerGCNLayer_1374389534964
// MI455X (gfx1250) — compile-verified
//
#include <hip/hip_runtime.h>
#include <hip/hip_bf16.h>
#include <cstdint>
#include <cstddef>

// ---------------------------------------------------------------------------
// TransformerConv-style graph attention layer for MI455X (gfx1250, wave32).
//
// Phases:
//   1) q/k/v/skip node projections : fp32 WMMA (V_WMMA_F32_16X16X4_F32),
//      B-fragments register-cached across 8 M-tiles per wave
//   2) edge logits  alpha[e][h]    : wave-per-edge; We/be staged in LDS
//      (async global->LDS via GLOBAL_LOAD_ASYNC_TO_LDS_B128 when available)
//   3) segment softmax (max + sum) : float atomic max (CAS) + atomicAdd
//   4) weighted scatter-add        : wave-per-edge, f32 global atomics
//   5) skip + relu                 : elementwise
//
// Workspace layout (floats): q | k | v | agg | alpha(ExH) | amax(NxH) | asum(NxH)
// ---------------------------------------------------------------------------

typedef float v2f __attribute__((ext_vector_type(2)));
typedef float v8f __attribute__((ext_vector_type(8)));
typedef int   v4i __attribute__((ext_vector_type(4)));

#define D_IN_   128
#define D_EDGE_ 32
#define HC_     256   // HEADS * D_HEAD
#define HEADS_  4
#define DHEAD_  64
#define MT_     8     // M-tiles per wave in the WMMA GEMM

#if __has_builtin(__builtin_amdgcn_global_load_async_to_lds_b128) && \
    __has_builtin(__builtin_amdgcn_s_wait_asynccnt)
#define USE_ASYNC_LDS 1
#endif

// ------------------------------ init ---------------------------------------
__global__ void init_buffers(float* __restrict__ agg,
                             float* __restrict__ amax,
                             float* __restrict__ asum,
                             int nAgg, int nNH) {
  int i = blockIdx.x * blockDim.x + threadIdx.x;
  if (i < nAgg) agg[i] = 0.0f;
  if (i < nNH) {
    amax[i] = -__builtin_inff();
    asum[i] = 0.0f;
  }
}

// --------------------- node projection: x @ W + b (WMMA) -------------------
// One wave owns a 16-wide N slice and 8 consecutive M-tiles. The 32 B-frags
// (full K=128) are loaded once into 64 VGPRs and reused for all 8 tiles.
// A-frag (16x4 f32): lanes 0-15 -> M=lane, K = {kk, kk+1}; lanes 16-31 -> K = {kk+2, kk+3}.
// B-frag (4x16 f32): lanes 0-15 -> N=lane, K = {kk, kk+1}; lanes 16-31 -> K = {kk+2, kk+3}.
// C/D (16x16 f32):   VGPR r -> M = r + 8*(lane>=16), N = lane&15.
__global__ __launch_bounds__(32) void node_proj_wmma(
    const float* __restrict__ x, const float* __restrict__ W,
    const float* __restrict__ bias, float* __restrict__ out, int M) {
  const int lane = threadIdx.x & 31;
  const int half = lane >> 4;   // 0 or 1
  const int l15  = lane & 15;
  const int n0    = blockIdx.y * 16;
  const int mbase = blockIdx.x * (16 * MT_);

  // Register-cache all B fragments for K = 0..127 (64 VGPRs).
  v2f bfrag[D_IN_ / 4];
  const float* bcol = W + (size_t)(2 * half) * HC_ + n0 + l15;
#pragma unroll
  for (int s = 0; s < D_IN_ / 4; ++s) {
    bfrag[s].x = bcol[(size_t)(4 * s) * HC_];
    bfrag[s].y = bcol[(size_t)(4 * s) * HC_ + HC_];
  }
  const float bn = bias[n0 + l15];

  for (int t = 0; t < MT_; ++t) {
    const int m0 = mbase + t * 16;
    if (m0 >= M) break;
    int mrow = m0 + l15;
    if (mrow >= M) mrow = M - 1;                   // clamp edge-tile loads
    const float* arow = x + (size_t)mrow * D_IN_ + 2 * half;

    v8f acc = {};
#pragma unroll
    for (int s = 0; s < D_IN_ / 4; ++s) {
      v2f a;
      a.x = arow[4 * s];
      a.y = arow[4 * s + 1];
      acc = __builtin_amdgcn_wmma_f32_16x16x4_f32(
          /*neg_a=*/false, a, /*neg_b=*/false, bfrag[s],
          /*c_mod=*/(short)0, acc, /*reuse_a=*/false, /*reuse_b=*/false);
    }

#pragma unroll
    for (int r = 0; r < 8; ++r) {
      const int m = m0 + 8 * half + r;
      if (m < M) out[(size_t)m * HC_ + n0 + l15] = acc[r] + bn;
    }
  }
}

// ------------------------- float atomic max (CAS) ---------------------------
__device__ inline void atomicMaxFloat(float* addr, float val) {
  unsigned int* ai = (unsigned int*)addr;
  unsigned int old = *ai;
  while (true) {
    float f = __uint_as_float(old);
    if (f >= val) break;
    unsigned int assumed = old;
    old = atomicCAS(ai, assumed, __float_as_uint(val));
    if (old == assumed) break;
  }
}

// ---------------- stage We (32 KB) + be (1 KB) into LDS ---------------------
__device__ inline void stage_we_lds(const float* __restrict__ We,
                                    const float* __restrict__ be,
                                    float* lds_we, float* lds_be) {
  const int tid = threadIdx.x;
#ifdef USE_ASYNC_LDS
  // CDNA5 async global->LDS: 2048 x 16-byte chunks, no VGPR round-trip.
  for (int c = tid; c < (D_EDGE_ * HC_) / 4; c += 256) {
    __builtin_amdgcn_global_load_async_to_lds_b128(
        (v4i*)(We + 4 * c), (v4i*)(lds_we + 4 * c), 0, 0);
  }
  for (int i = tid; i < HC_; i += 256) lds_be[i] = be[i];
  __builtin_amdgcn_s_wait_asynccnt(0);
#else
  for (int i = tid; i < D_EDGE_ * HC_; i += 256) lds_we[i] = We[i];
  for (int i = tid; i < HC_; i += 256) lds_be[i] = be[i];
#endif
  __syncthreads();
}

// ------------------- edge logits: alpha = <q_i, k_j + e> / 8 ----------------
// One wave (32 lanes) per edge. Lane l owns columns [8l, 8l+8) -> head = l>>3.
// edge_attr (32 floats) broadcast across the wave via __shfl (wave32 fit);
// We/be come from LDS (amortized over the block's 8 edges).
__global__ __launch_bounds__(256) void edge_alpha_kernel(
    const long long* __restrict__ ei, const float* __restrict__ ea,
    const float* __restrict__ We, const float* __restrict__ be,
    const float* __restrict__ q, const float* __restrict__ k,
    float* __restrict__ alpha, float* __restrict__ amax, int E) {
  __shared__ float lds_we[D_EDGE_ * HC_];
  __shared__ float lds_be[HC_];
  stage_we_lds(We, be, lds_we, lds_be);

  const int wave = threadIdx.x >> 5;
  const int lane = threadIdx.x & 31;
  const long long e = (long long)blockIdx.x * 8 + wave;
  if (e < E) {
    const int src = (int)ei[e];
    const int dst = (int)ei[(long long)E + e];

    const float ea_l = ea[e * D_EDGE_ + lane];    // one attr per lane

    const int c0 = lane * 8;
    float ep[8];
#pragma unroll
    for (int u = 0; u < 8; ++u) ep[u] = lds_be[c0 + u];
#pragma unroll
    for (int j = 0; j < D_EDGE_; ++j) {
      const float aj = __shfl(ea_l, j, 32);
      const float* wr = lds_we + j * HC_ + c0;
#pragma unroll
      for (int u = 0; u < 8; ++u) ep[u] = fmaf(aj, wr[u], ep[u]);
    }

    const float* qrow = q + (size_t)dst * HC_ + c0;
    const float* krow = k + (size_t)src * HC_ + c0;
    float p = 0.0f;
#pragma unroll
    for (int u = 0; u < 8; ++u) p = fmaf(qrow[u], krow[u] + ep[u], p);

    // reduce within each 8-lane head group
    p += __shfl_xor(p, 1, 32);
    p += __shfl_xor(p, 2, 32);
    p += __shfl_xor(p, 4, 32);
    p *= 0.125f;                                   // 1/sqrt(D_HEAD=64)

    if ((lane & 7) == 0) {
      const int h = lane >> 3;
      alpha[e * HEADS_ + h] = p;
      atomicMaxFloat(&amax[(size_t)dst * HEADS_ + h], p);
    }
  }
}

// ----------------- softmax numerator: ex = exp(a - amax), sum ---------------
__global__ void edge_softmax_num(const long long* __restrict__ ei,
                                 float* __restrict__ alpha,
                                 const float* __restrict__ amax,
                                 float* __restrict__ asum, int E) {
  const int i = blockIdx.x * blockDim.x + threadIdx.x;  // over E*HEADS
  if (i >= E * HEADS_) return;
  const int e = i >> 2;
  const int h = i & 3;
  const int dst = (int)ei[(long long)E + e];
  const float m = amax[(size_t)dst * HEADS_ + h];       // finite: dst has >=1 edge
  const float ex = __expf(alpha[i] - m);
  alpha[i] = ex;                                        // overwrite with numerator
  atomicAdd(&asum[(size_t)dst * HEADS_ + h], ex);
}

// ----------- aggregation: agg[dst] += attn * (v[src] + e_proj) --------------
__global__ __launch_bounds__(256) void edge_aggregate(
    const long long* __restrict__ ei, const float* __restrict__ ea,
    const float* __restrict__ We, const float* __restrict__ be,
    const float* __restrict__ v, const float* __restrict__ alpha,
    const float* __restrict__ asum, float* __restrict__ agg, int E) {
  __shared__ float lds_we[D_EDGE_ * HC_];
  __shared__ float lds_be[HC_];
  stage_we_lds(We, be, lds_we, lds_be);

  const int wave = threadIdx.x >> 5;
  const int lane = threadIdx.x & 31;
  const long long e = (long long)blockIdx.x * 8 + wave;
  if (e < E) {
    const int src = (int)ei[e];
    const int dst = (int)ei[(long long)E + e];

    const float ea_l = ea[e * D_EDGE_ + lane];

    const int c0 = lane * 8;
    const int h = lane >> 3;
    const float attn = alpha[e * HEADS_ + h] / asum[(size_t)dst * HEADS_ + h];

    float ep[8];
#pragma unroll
    for (int u = 0; u < 8; ++u) ep[u] = lds_be[c0 + u];
#pragma unroll
    for (int j = 0; j < D_EDGE_; ++j) {
      const float aj = __shfl(ea_l, j, 32);
      const float* wr = lds_we + j * HC_ + c0;
#pragma unroll
      for (int u = 0; u < 8; ++u) ep[u] = fmaf(aj, wr[u], ep[u]);
    }

    const float* vrow = v + (size_t)src * HC_ + c0;
    float* orow = agg + (size_t)dst * HC_ + c0;
#pragma unroll
    for (int u = 0; u < 8; ++u) {
      atomicAdd(&orow[u], attn * (vrow[u] + ep[u]));
    }
  }
}

// --------------------------- out = relu(skip + agg) -------------------------
__global__ void finalize_kernel(float* __restrict__ out,
                                const float* __restrict__ agg, int n) {
  const int i = blockIdx.x * blockDim.x + threadIdx.x;
  if (i >= n) return;
  const float o = out[i] + agg[i];
  out[i] = o > 0.0f ? o : 0.0f;
}

// ---------------------------------------------------------------------------
extern "C" void kernel_launch(void* const* d_in, const int* in_sizes, int n_in,
                              void* d_out, int out_size, void* d_ws, size_t ws_size,
                              hipStream_t stream) {
  const float*     x   = (const float*)d_in[0];
  const long long* ei  = (const long long*)d_in[1];   // int64 edge_index [2, E]
  const float*     ea  = (const float*)d_in[2];
  const float*     Wq  = (const float*)d_in[3];
  const float*     bq  = (const float*)d_in[4];
  const float*     Wk  = (const float*)d_in[5];
  const float*     bk  = (const float*)d_in[6];
  const float*     Wv  = (const float*)d_in[7];
  const float*     bv  = (const float*)d_in[8];
  const float*     We  = (const float*)d_in[9];
  const float*     be  = (const float*)d_in[10];
  const float*     Ws  = (const float*)d_in[11];
  const float*     bs  = (const float*)d_in[12];

  const int N = in_sizes[0] / D_IN_;   // 50000
  const int E = in_sizes[1] / 2;       // 800000

  // Workspace carve-up (floats). Total ~220 MB.
  float* ws    = (float*)d_ws;
  const size_t NQ = (size_t)N * HC_;
  float* q     = ws;
  float* k     = q + NQ;
  float* v     = k + NQ;
  float* agg   = v + NQ;
  float* alpha = agg + NQ;                 // E * HEADS
  float* amax  = alpha + (size_t)E * HEADS_;
  float* asum  = amax + (size_t)N * HEADS_;

  float* out = (float*)d_out;

  // 1) init (d_ws is poisoned by the harness)
  {
    const int nAgg = (int)NQ;
    const int nNH  = N * HEADS_;
    init_buffers<<<(nAgg + 255) / 256, 256, 0, stream>>>(agg, amax, asum,
                                                         nAgg, nNH);
  }

  // 2) node projections via fp32 WMMA: q, k, v -> ws; skip -> d_out
  {
    dim3 grid((N + 16 * MT_ - 1) / (16 * MT_), HC_ / 16);
    node_proj_wmma<<<grid, 32, 0, stream>>>(x, Wq, bq, q, N);
    node_proj_wmma<<<grid, 32, 0, stream>>>(x, Wk, bk, k, N);
    node_proj_wmma<<<grid, 32, 0, stream>>>(x, Wv, bv, v, N);
    node_proj_wmma<<<grid, 32, 0, stream>>>(x, Ws, bs, out, N);
  }

  // 3) attention logits + segment max
  {
    const int blocks = (E + 7) / 8;     // 8 waves (edges) per 256-thread block
    edge_alpha_kernel<<<blocks, 256, 0, stream>>>(ei, ea, We, be, q, k,
                                                  alpha, amax, E);
  }

  // 4) exp + segment sum
  {
    const int total = E * HEADS_;
    edge_softmax_num<<<(total + 255) / 256, 256, 0, stream>>>(ei, alpha, amax,
                                                              asum, E);
  }

  // 5) weighted scatter-add
  {
    const int blocks = (E + 7) / 8;
    edge_aggregate<<<blocks, 256, 0, stream>>>(ei, ea, We, be, v, alpha, asum,
                                               agg, E);
  }

  // 6) skip + relu
  {
    const int total = (int)NQ;
    finalize_kernel<<<(total + 255) / 256, 256, 0, stream>>>(out, agg, total);
  }
}